// CRFLossOld_49675591746148
// MI455X (gfx1250) — compile-verified
//
#include <hip/hip_runtime.h>
#include <hip/hip_bf16.h>
#include <math.h>

#define CRF_B 1024
#define CRF_T 512
#define CRF_L 62
#define CRF_K 64
#define CRF_SMALL (-1000.0f)

typedef float v2f __attribute__((ext_vector_type(2)));
typedef float v8f __attribute__((ext_vector_type(8)));

// ---------------------------------------------------------------------------
// Kernel 1: init.  E = exp(transitions) (64x64), c_j = lse_i(b_s_i + trans_ij)
// (exact log-space step-1 constant, batch independent), zero the output.
// ---------------------------------------------------------------------------
__global__ void crf_init_kernel(const float* __restrict__ trans,
                                float* __restrict__ E,
                                float* __restrict__ cvec,
                                float* __restrict__ out) {
  const int tid = threadIdx.x;  // one block of 256
  if (tid == 0) out[0] = 0.0f;
  for (int i = tid; i < CRF_K * CRF_K; i += 256) E[i] = __expf(trans[i]);
  if (tid < CRF_K) {
    const int j = tid;
    float m = -3.0e38f;
    for (int i = 0; i < CRF_K; ++i) {
      float s = trans[i * CRF_K + j] + ((i == CRF_L) ? 0.0f : CRF_SMALL);
      m = fmaxf(m, s);
    }
    float sum = 0.0f;
    for (int i = 0; i < CRF_K; ++i) {
      float s = trans[i * CRF_K + j] + ((i == CRF_L) ? 0.0f : CRF_SMALL);
      sum += __expf(s - m);
    }
    cvec[j] = m + __logf(sum);
  }
}

// ---------------------------------------------------------------------------
// Kernel 2: forward recursion via f32 WMMA.
// Block = 128 threads (4 waves), handles 16 batch rows.  Wave w owns output
// columns [16w, 16w+16).  E column slab lives in 16 v2f B-fragments.
// Per step: P = exp(alpha - rowmax) -> C = P @ E via 16 v_wmma_f32_16x16x4_f32
// split over 4 independent accumulators (chain depth 4), then
// alpha' = log(C) + rowmax + obs.  Double-buffered LDS alpha, 1 barrier/step.
// ---------------------------------------------------------------------------
__launch_bounds__(128, 1)
__global__ void crf_forward_kernel(const float* __restrict__ pred,
                                   const int* __restrict__ seq_len,
                                   const float* __restrict__ E,
                                   const float* __restrict__ cvec,
                                   float* __restrict__ out) {
  __shared__ float alpha[2][16][CRF_K];
  __shared__ float rowres[16];

  const int tid   = threadIdx.x;
  const int wid   = tid >> 5;
  const int lane  = tid & 31;
  const int hi    = lane >> 4;        // half-wave: selects K pair {0,1} vs {2,3}
  const int lrow  = lane & 15;        // A-matrix row this lane feeds
  const int col   = wid * 16 + lrow;  // global output column (0..63)
  const int bbase = blockIdx.x * 16;

  // B fragments: E[k][col] for k = 4*kc + 2*hi + {0,1}.  Loaded once.
  v2f bfrag[16];
#pragma unroll
  for (int kc = 0; kc < 16; ++kc) {
    bfrag[kc].x = E[(4 * kc + 2 * hi + 0) * CRF_K + col];
    bfrag[kc].y = E[(4 * kc + 2 * hi + 1) * CRF_K + col];
  }

  // seq_len + hoisted pred row offsets for the 8 C rows this lane owns.
  int slen[8];
  size_t rowoff[8];
#pragma unroll
  for (int v = 0; v < 8; ++v) {
    const int r = hi * 8 + v;
    slen[v]   = seq_len[bbase + r];
    rowoff[v] = ((size_t)(bbase + r) * CRF_T) * CRF_L + col;
  }

  int tmax = 0;
  for (int r = 0; r < 16; ++r) tmax = max(tmax, seq_len[bbase + r]);
  tmax += 1;  // capture times are seq_len+1 in [2, T+1]

  const int scap = (wid == 0 && lane < 16) ? seq_len[bbase + lane] : -0x40000000;

  // alpha_1[b][k] = obs(t=1) + c_k   (step 1 done exactly in log space)
  {
    const int e0  = tid * 8;
    const int row = e0 >> 6;
    const int k0  = e0 & 63;
#pragma unroll
    for (int q = 0; q < 8; ++q) {
      const int k = k0 + q;
      const float ob = (k < CRF_L)
          ? pred[((size_t)(bbase + row) * CRF_T) * CRF_L + k] : CRF_SMALL;
      alpha[0][row][k] = ob + cvec[k];
    }
  }
  __syncthreads();

  int cur = 0;
  for (int t = 2; t <= tmax; ++t) {
    const int nxt = cur ^ 1;

    // ---- build A fragments: each lane reads exactly its 2 floats/chunk,
    //      partial row max, combine with half-wave partner via shfl_xor ----
    float m = -3.0e38f;
    v2f araw[16];
#pragma unroll
    for (int kc = 0; kc < 16; ++kc) {
      const v2f q = *(const v2f*)&alpha[cur][lrow][4 * kc + 2 * hi];
      araw[kc] = q;
      m = fmaxf(m, fmaxf(q.x, q.y));
    }
    m = fmaxf(m, __shfl_xor(m, 16, 32));  // full row max (rows held by l and l+16)

    v2f afrag[16];
#pragma unroll
    for (int kc = 0; kc < 16; ++kc) {
      afrag[kc].x = __expf(araw[kc].x - m);
      afrag[kc].y = __expf(araw[kc].y - m);
    }

    // ---- C(16x16) = P(16x64) @ E(64x64) column slab, f32 tensor op.
    //      4 independent accumulators -> dependent-WMMA chain depth 4. ----
    v8f acc0 = {}, acc1 = {}, acc2 = {}, acc3 = {};
#pragma unroll
    for (int g = 0; g < 4; ++g) {
      acc0 = __builtin_amdgcn_wmma_f32_16x16x4_f32(false, afrag[4*g+0], false, bfrag[4*g+0], (short)0, acc0, false, false);
      acc1 = __builtin_amdgcn_wmma_f32_16x16x4_f32(false, afrag[4*g+1], false, bfrag[4*g+1], (short)0, acc1, false, false);
      acc2 = __builtin_amdgcn_wmma_f32_16x16x4_f32(false, afrag[4*g+2], false, bfrag[4*g+2], (short)0, acc2, false, false);
      acc3 = __builtin_amdgcn_wmma_f32_16x16x4_f32(false, afrag[4*g+3], false, bfrag[4*g+3], (short)0, acc3, false, false);
    }
    const v8f c = (acc0 + acc1) + (acc2 + acc3);

    // ---- alpha'[r][col] = log(C) + rowmax(r) + obs(t, r, col) ----
    const bool lastT = (t == CRF_T + 1);
#pragma unroll
    for (int v = 0; v < 8; ++v) {
      const int r = hi * 8 + v;
      const float mr = __shfl(m, r, 32);  // lane r holds rowmax of row r
      float ob;
      if (col < CRF_L) {
        ob = lastT ? 0.0f : pred[rowoff[v] + (size_t)(t - 1) * CRF_L];
        if (t < CRF_T)  // prefetch next step's emission row
          __builtin_prefetch(&pred[rowoff[v] + (size_t)t * CRF_L], 0, 0);
      } else {
        ob = CRF_SMALL;
      }
      if (t == slen[v] + 1)  // e_s injection at capture time
        ob += (col < CRF_L) ? CRF_SMALL : ((col == CRF_L) ? 0.0f : 1000.0f);
      alpha[nxt][r][col] = __logf(c[v]) + mr + ob;
    }
    __syncthreads();

    // ---- capture lse(alpha_{seq_len+1}) per finished row (wave 0) ----
    if (scap + 1 == t) {
      const float4* rp4 = (const float4*)&alpha[nxt][lane][0];
      float m2 = -3.0e38f;
#pragma unroll
      for (int k = 0; k < 16; ++k) {
        const float4 q = rp4[k];
        m2 = fmaxf(fmaxf(fmaxf(q.x, q.y), fmaxf(q.z, q.w)), m2);
      }
      float s = 0.0f;
#pragma unroll
      for (int k = 0; k < 16; ++k) {
        const float4 q = rp4[k];
        s += __expf(q.x - m2) + __expf(q.y - m2) +
             __expf(q.z - m2) + __expf(q.w - m2);
      }
      rowres[lane] = m2 + __logf(s);
    }
    cur = nxt;
  }
  __syncthreads();

  if (tid == 0) {
    float s = 0.0f;
    for (int r = 0; r < 16; ++r) s += rowres[r];
    atomicAdd(out, s);
  }
}

// ---------------------------------------------------------------------------
// Kernel 3: gold-path (real) score, subtracted from the output.
// One wave per batch element; shuffle reduction; one f32 atomic per wave.
// ---------------------------------------------------------------------------
__global__ void crf_gold_kernel(const float* __restrict__ pred,
                                const int* __restrict__ ref,
                                const int* __restrict__ seq_len,
                                const float* __restrict__ trans,
                                float* __restrict__ out) {
  const int b = blockIdx.x * (blockDim.x >> 5) + (threadIdx.x >> 5);
  const int lane = threadIdx.x & 31;
  if (b >= CRF_B) return;
  const int n = seq_len[b];
  const int* rb = ref + (size_t)b * CRF_T;
  const float* pb = pred + (size_t)b * CRF_T * CRF_L;
  float s = 0.0f;
  for (int t = lane; t < n; t += 32)       // emissions
    s += pb[(size_t)t * CRF_L + rb[t]];
  for (int t = lane; t <= n; t += 32) {    // transitions along gold path
    const int p = (t == 0) ? CRF_L : rb[t - 1];
    const int q = (t == n) ? (CRF_L + 1) : rb[t];
    s += trans[p * CRF_K + q];
  }
  for (int off = 16; off; off >>= 1) s += __shfl_down(s, off, 32);
  if (lane == 0) atomicAdd(out, -s);
}

// ---------------------------------------------------------------------------
extern "C" void kernel_launch(void* const* d_in, const int* in_sizes, int n_in,
                              void* d_out, int out_size, void* d_ws, size_t ws_size,
                              hipStream_t stream) {
  (void)in_sizes; (void)n_in; (void)out_size; (void)ws_size;
  const float* pred    = (const float*)d_in[0];   // (B,T,L) f32
  const int*   ref     = (const int*)d_in[1];     // (B,T) i32
  const int*   seqlen  = (const int*)d_in[2];     // (B,) i32
  const float* trans   = (const float*)d_in[3];   // (K,K) f32
  float* out = (float*)d_out;

  float* E    = (float*)d_ws;            // 64*64 f32
  float* cvec = E + CRF_K * CRF_K;       // 64 f32

  crf_init_kernel<<<1, 256, 0, stream>>>(trans, E, cvec, out);
  crf_forward_kernel<<<CRF_B / 16, 128, 0, stream>>>(pred, seqlen, E, cvec, out);
  crf_gold_kernel<<<CRF_B / 4, 128, 0, stream>>>(pred, ref, seqlen, trans, out);
}